// Mask2FormerCriterion_83648783057219
// MI455X (gfx1250) — compile-verified
//
#include <hip/hip_runtime.h>
#include <math.h>

#define B_    2
#define Q_    100
#define C_    4
#define M_    32
#define H_    384
#define W_    384
#define HW_   (H_*W_)
#define N_    (B_*M_)          // 64 instances
#define CH_   36               // chunks per instance (HW = 36*4096)
#define CHSZ_ 4096
#define NP2_  262144           // 2^18 >= HW, bitonic pad
#define LOGN2_ 18
#define LBS_  2048             // LDS bitonic block span (16KB of float2)
#define EPSF  1e-6f
#define NQTY_ 10               // reduced quantities

#if defined(__has_builtin)
#if __has_builtin(__builtin_amdgcn_wmma_f32_16x16x4_f32)
#define HAVE_WMMA_F32X4 1
#endif
#endif

typedef float v2f __attribute__((ext_vector_type(2)));
typedef float v8f __attribute__((ext_vector_type(8)));

// ---------------- helpers ----------------
__device__ __forceinline__ float sigf(float x){ return 1.f/(1.f+expf(-x)); }
__device__ __forceinline__ float bcef(float x, float g){
  return fmaxf(x,0.f) - x*g + log1pf(expf(-fabsf(x)));
}
__device__ __forceinline__ float blockReduce256(float v, float* sm){
  int t = threadIdx.x;
  sm[t] = v; __syncthreads();
  #pragma unroll
  for (int off=128; off>0; off>>=1){ if (t<off) sm[t]+=sm[t+off]; __syncthreads(); }
  float r = sm[0]; __syncthreads();
  return r;
}

// ---------------- focal classification loss ----------------
__global__ void k_focal(const float* __restrict__ logits, const int* __restrict__ glab,
                        const int* __restrict__ pidx, float* __restrict__ lcls){
  __shared__ int   tgt[Q_];
  __shared__ float sr[128];
  __shared__ float sn[128];
  int b = blockIdx.x, t = threadIdx.x;
  for (int i=t;i<Q_;i+=128) tgt[i]=0;
  __syncthreads();
  if (t < M_) tgt[pidx[b*M_+t]] = glab[b*M_+t];
  __syncthreads();
  float acc=0.f, nf=0.f;
  for (int i=t;i<Q_*C_;i+=128){
    int q=i/C_, c=i%C_;
    float x  = logits[(b*Q_+q)*C_+c];
    float oh = (tgt[q]==c)?1.f:0.f;
    float pr = sigf(x);
    float ce = bcef(x,oh);
    float pt = pr*oh + (1.f-pr)*(1.f-oh);
    float om = 1.f-pt;
    acc += 0.25f*om*om*ce;
  }
  for (int i=t;i<Q_;i+=128) nf += (tgt[i] >= 1)?1.f:0.f;
  sr[t]=acc; sn[t]=nf; __syncthreads();
  for (int off=64; off>0; off>>=1){
    if (t<off){ sr[t]+=sr[t+off]; sn[t]+=sn[t+off]; } __syncthreads();
  }
  if (t==0) lcls[b] = sr[0]/fmaxf(sn[0],1.f);
}

// ---------------- elementwise fills ----------------
__global__ void k_sigmoid_fill(const float* __restrict__ pm_all, const int* __restrict__ pidx,
                               float* __restrict__ ps){
  int gid = blockIdx.x*256 + threadIdx.x;
  int n = gid / HW_, i = gid % HW_;
  int b = n / M_;
  const float* pm = pm_all + (size_t)(b*Q_ + pidx[n]) * HW_;
  ps[gid] = sigf(pm[i]);
}

__global__ void k_gather_gm(const float* __restrict__ gm_all, const int* __restrict__ gidx,
                            float* __restrict__ dst){
  int gid = blockIdx.x*256 + threadIdx.x;
  int n = gid / HW_, i = gid % HW_;
  int b = n / M_;
  dst[gid] = gm_all[(size_t)(b*M_ + gidx[n])*HW_ + i];
}

__global__ void k_copy(const float* __restrict__ s, float* __restrict__ d){
  int gid = blockIdx.x*256 + threadIdx.x; d[gid] = s[gid];
}
__global__ void k_zero(float* __restrict__ d){
  int gid = blockIdx.x*256 + threadIdx.x; d[gid] = 0.f;
}

// ---------------- stage-1 chunk reductions ----------------
__global__ void k_base_stage1(const float* __restrict__ pm_all, const float* __restrict__ gm_all,
                              const int* __restrict__ pidx, const int* __restrict__ gidx,
                              float* __restrict__ part){
  __shared__ float sm[256];
  int n = blockIdx.x / CH_, c = blockIdx.x % CH_;
  int b = n / M_, t = threadIdx.x;
  const float* pm = pm_all + (size_t)(b*Q_ + pidx[n]) * HW_;
  const float* gm = gm_all + (size_t)(b*M_ + gidx[n]) * HW_;
  int base = c*CHSZ_;
  float sce=0.f, sint=0.f, sps=0.f, sgs=0.f;
  for (int s=0; s<CHSZ_/256; ++s){
    int i = base + s*256 + t;
    float x = pm[i], g = gm[i];
    float sp = sigf(x);
    sce += bcef(x,g); sint += sp*g; sps += sp; sgs += g;
  }
  float r;
  r = blockReduce256(sce , sm); if(t==0) part[0*N_*CH_ + n*CH_ + c] = r;
  r = blockReduce256(sint, sm); if(t==0) part[1*N_*CH_ + n*CH_ + c] = r;
  r = blockReduce256(sps , sm); if(t==0) part[2*N_*CH_ + n*CH_ + c] = r;
  r = blockReduce256(sgs , sm); if(t==0) part[3*N_*CH_ + n*CH_ + c] = r;
}

// ---------------- boundary loss ----------------
__global__ void k_edges(const float* __restrict__ gm_all, const int* __restrict__ gidx,
                        float* __restrict__ edge){
  int gid = blockIdx.x*256 + threadIdx.x;
  int n = gid / HW_, i = gid % HW_;
  int b = n / M_;
  const float* gm = gm_all + (size_t)(b*M_ + gidx[n])*HW_;
  int y = i / W_, x = i % W_;
  int ym = max(y-1,0), yp = min(y+1,H_-1), xm = max(x-1,0), xp = min(x+1,W_-1);
  float lap = gm[ym*W_+x] + gm[yp*W_+x] + gm[y*W_+xm] + gm[y*W_+xp] - 4.f*gm[y*W_+x];
  edge[gid] = (fabsf(lap) > 0.1f) ? 1.f : 0.f;
}

__global__ void k_bnd_stage1(const float* __restrict__ pm_all, const float* __restrict__ gm_all,
                             const int* __restrict__ pidx, const int* __restrict__ gidx,
                             const float* __restrict__ edge, float* __restrict__ part){
  __shared__ float sm[256];
  int n = blockIdx.x / CH_, c = blockIdx.x % CH_;
  int b = n / M_, t = threadIdx.x;
  const float* pm = pm_all + (size_t)(b*Q_ + pidx[n]) * HW_;
  const float* gm = gm_all + (size_t)(b*M_ + gidx[n]) * HW_;
  const float* ed = edge + (size_t)n*HW_;
  int base = c*CHSZ_;
  float s0=0.f, s1=0.f;
  for (int s=0; s<CHSZ_/256; ++s){
    int i = base + s*256 + t;
    int y = i / W_, x = i % W_;
    float bd = 0.f;
    for (int dy=-1; dy<=1; ++dy){
      int yy = y+dy; if (yy<0 || yy>=H_) continue;
      for (int dx=-1; dx<=1; ++dx){
        int xx = x+dx; if (xx<0 || xx>=W_) continue;
        bd = fmaxf(bd, ed[yy*W_+xx]);
      }
    }
    float w  = 1.f + 4.f*bd;
    float ce = bcef(pm[i], gm[i]);
    s0 += w*ce; s1 += w;
  }
  float r;
  r = blockReduce256(s0, sm); if(t==0) part[4*N_*CH_ + n*CH_ + c] = r;
  r = blockReduce256(s1, sm); if(t==0) part[5*N_*CH_ + n*CH_ + c] = r;
}

// ---------------- fused soft-skeleton iteration (erode + open + skel accumulate) ----------------
// One block = one 32x32 output tile. img tile loaded with halo 2 (+inf outside image so
// min-pool excludes OOB, matching reduce_window SAME with -inf init on the negated image).
// Eroded ring (34x34) kept in LDS with -inf at OOB so the opening max-pool excludes it.
__global__ void k_skel_iter(const float* __restrict__ img, float* __restrict__ erodedOut,
                            float* __restrict__ skel){
  __shared__ float simg[36][36];
  __shared__ float ser[34][34];
  const int TX = W_/32, TY = H_/32;
  int bid = blockIdx.x;
  int n  = bid / (TX*TY);
  int tb = bid % (TX*TY);
  int ty0 = (tb / TX)*32, tx0 = (tb % TX)*32;
  const float* im = img + (size_t)n*HW_;
  int t = threadIdx.x;

  for (int idx=t; idx<36*36; idx+=256){
    int ly = idx/36, lx = idx%36;
    int y = ty0 + ly - 2, x = tx0 + lx - 2;
    float v = 3.4e38f;
    if (y>=0 && y<H_ && x>=0 && x<W_) v = im[y*W_+x];
    simg[ly][lx] = v;
  }
  __syncthreads();

  for (int idx=t; idx<34*34; idx+=256){
    int ly = idx/34, lx = idx%34;
    int y = ty0 + ly - 1, x = tx0 + lx - 1;
    float v;
    if (y>=0 && y<H_ && x>=0 && x<W_){
      v = 3.4e38f;
      #pragma unroll
      for (int dy=0; dy<3; ++dy)
        #pragma unroll
        for (int dx=0; dx<3; ++dx)
          v = fminf(v, simg[ly+dy][lx+dx]);
    } else v = -3.4e38f;
    ser[ly][lx] = v;
  }
  __syncthreads();

  for (int idx=t; idx<32*32; idx+=256){
    int ly = idx/32, lx = idx%32;
    int y = ty0+ly, x = tx0+lx;
    float er = ser[ly+1][lx+1];
    float mx = -3.4e38f;
    #pragma unroll
    for (int dy=0; dy<3; ++dy)
      #pragma unroll
      for (int dx=0; dx<3; ++dx)
        mx = fmaxf(mx, ser[ly+dy][lx+dx]);
    size_t gi = (size_t)n*HW_ + (size_t)y*W_ + x;
    erodedOut[gi] = er;
    skel[gi] += fmaxf(simg[ly+2][lx+2] - mx, 0.f);
  }
}

__global__ void k_skel_stage1(const float* __restrict__ skel, const float* __restrict__ gm_all,
                              const int* __restrict__ gidx, const float* __restrict__ psbuf,
                              int mode, float* __restrict__ partout){
  __shared__ float sm[256];
  int n = blockIdx.x / CH_, c = blockIdx.x % CH_;
  int b = n / M_, t = threadIdx.x;
  const float* sk = skel + (size_t)n*HW_;
  const float* other = (mode == 0)
      ? (gm_all + (size_t)(b*M_ + gidx[n])*HW_)
      : (psbuf + (size_t)n*HW_);
  int base = c*CHSZ_;
  float s0=0.f, s1=0.f;
  for (int s=0; s<CHSZ_/256; ++s){
    int i = base + s*256 + t;
    float sv = fminf(fmaxf(sk[i], 0.f), 1.f);
    s0 += sv*other[i]; s1 += sv;
  }
  float r;
  r = blockReduce256(s0, sm); if(t==0) partout[            n*CH_ + c] = r;
  r = blockReduce256(s1, sm); if(t==0) partout[N_*CH_ +    n*CH_ + c] = r;
}

// ---------------- WMMA chunk-partial reduction: row sums of [64 x 36] via A x ones ----------------
__global__ void k_wmma_reduce(const float* __restrict__ part, float* __restrict__ red){
  int q  = blockIdx.x >> 2;        // quantity
  int rg = blockIdx.x & 3;         // 16-row group (64 rows total)
  int lane = threadIdx.x;          // 32 threads (one wave32)
  const float* P = part + (size_t)q*N_*CH_;
#ifdef HAVE_WMMA_F32X4
  v8f c = {};
  v2f bm; bm.x = 1.f; bm.y = 1.f;  // all-ones B: D[i][j] = sum_k A[i][k]
  int row   = rg*16 + (lane & 15);
  int khalf = (lane >> 4) << 1;    // lanes 16-31 hold K=2,3 of each k-quad
  for (int k = 0; k < CH_/4; ++k){
    v2f a;
    a.x = P[row*CH_ + 4*k + khalf];
    a.y = P[row*CH_ + 4*k + khalf + 1];
    c = __builtin_amdgcn_wmma_f32_16x16x4_f32(false, a, false, bm, (short)0, c, false, false);
  }
  if ((lane & 15) == 0){           // lane0 -> rows rg*16..+7, lane16 -> rows rg*16+8..+15
    int mb = rg*16 + ((lane>>4)<<3);
    #pragma unroll
    for (int r=0;r<8;++r) red[q*N_ + mb + r] = c[r];
  }
#else
  if (lane < 16){
    int row = rg*16 + lane;
    float s = 0.f;
    for (int k=0;k<CH_;++k) s += P[row*CH_ + k];
    red[q*N_ + row] = s;
  }
#endif
}

// ---------------- Lovasz hinge ----------------
__global__ void k_lov_fill(const float* __restrict__ pm_all, const float* __restrict__ gm_all,
                           const int* __restrict__ pidx, const int* __restrict__ gidx,
                           float2* __restrict__ buf){
  int gid = blockIdx.x*256 + threadIdx.x;
  int n = gid >> LOGN2_, i = gid & (NP2_-1);
  float2 v;
  if (i < HW_){
    int b = n / M_;
    float x = pm_all[(size_t)(b*Q_ + pidx[n])*HW_ + i];
    float g = gm_all[(size_t)(b*M_ + gidx[n])*HW_ + i];
    v.x = 1.f - (2.f*g - 1.f)*x;   // error
    v.y = g;
  } else {
    v.x = -1e30f; v.y = 0.f;       // pad: sorts last, relu(err)=0 -> no contribution
  }
  buf[gid] = v;
}

// global compare-exchange step (used only while j >= LBS_)
__global__ void k_bitonic_global(float2* __restrict__ buf, unsigned k, unsigned j){
  int gid = blockIdx.x*256 + threadIdx.x;
  int n = gid >> LOGN2_;
  unsigned i = (unsigned)(gid & (NP2_-1));
  unsigned p = i ^ j;
  if (p > i){
    size_t base = (size_t)n << LOGN2_;
    float2 a = buf[base + i];
    float2 b = buf[base + p];
    bool desc = ((i & k) == 0);    // descending overall
    bool sw = desc ? (a.x < b.x) : (a.x > b.x);
    if (sw){ buf[base + i] = b; buf[base + p] = a; }
  }
}

// LDS-resident tail: runs all steps j = min(k/2, LBS_/2) .. 1 for a 2048-element span
__global__ void k_bitonic_lds(float2* __restrict__ buf, unsigned k){
  __shared__ float2 sm[LBS_];
  int t = threadIdx.x;                               // 256
  size_t blockStart = (size_t)blockIdx.x * LBS_;     // NP2 % LBS == 0 -> span stays in one instance
  #pragma unroll
  for (int s=0; s<LBS_/256; ++s) sm[s*256+t] = buf[blockStart + s*256 + t];
  __syncthreads();
  unsigned ibase = (unsigned)(blockStart & (NP2_-1));
  unsigned jstart = (k>>1 < (unsigned)(LBS_/2)) ? (k>>1) : (unsigned)(LBS_/2);
  for (unsigned j = jstart; j > 0; j >>= 1){
    #pragma unroll
    for (int s=0; s<LBS_/512; ++s){
      unsigned p  = (unsigned)(s*256 + t);           // pair index 0..LBS/2-1
      unsigned lo = ((p & ~(j-1u)) << 1) | (p & (j-1u));
      unsigned hi = lo + j;
      float2 a = sm[lo], b = sm[hi];
      bool desc = (((ibase + lo) & k) == 0);
      bool sw = desc ? (a.x < b.x) : (a.x > b.x);
      if (sw){ sm[lo] = b; sm[hi] = a; }
    }
    __syncthreads();
  }
  #pragma unroll
  for (int s=0; s<LBS_/256; ++s) buf[blockStart + s*256 + t] = sm[s*256+t];
}

__global__ void k_lov_scan(const float2* __restrict__ buf, const float* __restrict__ gsum,
                           float* __restrict__ lov){
  __shared__ float sg[256];
  __shared__ float sr[256];
  int n = blockIdx.x, t = threadIdx.x;
  const float2* p = buf + ((size_t)n << LOGN2_);
  float gts = gsum[n];
  float carry = 0.f, acc = 0.f;
  for (int ch = 0; ch < NP2_/256; ++ch){
    int i = ch*256 + t;
    float2 v = p[i];
    sg[t] = v.y;
    __syncthreads();
    for (int off=1; off<256; off<<=1){      // Hillis-Steele inclusive scan
      float add = (t>=off) ? sg[t-off] : 0.f;
      __syncthreads();
      sg[t] += add;
      __syncthreads();
    }
    float cum = carry + sg[t];
    float fi  = (float)i;
    float jac = 1.f - (gts - cum)/(gts + fi + 1.f - cum + EPSF);
    float grad;
    if (i == 0) grad = jac;
    else {
      float cumP = cum - v.y;
      float jacP = 1.f - (gts - cumP)/(gts + fi - cumP + EPSF);
      grad = jac - jacP;
    }
    acc += fmaxf(v.x, 0.f) * grad;
    float tot = sg[255];
    __syncthreads();
    carry += tot;
  }
  sr[t] = acc; __syncthreads();
  for (int off=128; off>0; off>>=1){ if (t<off) sr[t]+=sr[t+off]; __syncthreads(); }
  if (t==0) lov[n] = sr[0];
}

// ---------------- final combine ----------------
__global__ void k_combine(const float* __restrict__ red, const float* __restrict__ lcls,
                          const float* __restrict__ lov, float* __restrict__ out){
  if (threadIdx.x || blockIdx.x) return;
  float total = 0.f;
  for (int b=0; b<B_; ++b){
    float sce=0.f, sdice=0.f, bnum=0.f, bden=0.f, scld=0.f, slov=0.f;
    for (int m=0; m<M_; ++m){
      int n = b*M_ + m;
      float ce   = red[0*N_+n], inter = red[1*N_+n];
      float psum = red[2*N_+n], gsum  = red[3*N_+n];
      sce  += ce;
      sdice += 1.f - 2.f*inter/(psum + gsum + EPSF);
      bnum += red[4*N_+n]; bden += red[5*N_+n];
      float tp = red[6*N_+n]/(red[7*N_+n] + EPSF);
      float ts = red[8*N_+n]/(red[9*N_+n] + EPSF);
      scld += 1.f - 2.f*tp*ts/(tp + ts + EPSF);
      slov += lov[n];
    }
    float l_bce  = sce / (float)(M_*HW_);
    float l_dice = sdice / (float)M_;
    float l_bnd  = bnum / (bden + EPSF);
    float l_cld  = scld / (float)M_;
    float l_lov  = slov / (float)M_;
    total += 2.f*lcls[b] + 5.f*l_bce + 5.f*l_dice + 2.f*l_bnd + 3.f*l_cld + 1.f*l_lov;
  }
  out[0] = total / (float)B_;
}

// ---------------- host launch ----------------
extern "C" void kernel_launch(void* const* d_in, const int* in_sizes, int n_in,
                              void* d_out, int out_size, void* d_ws, size_t ws_size,
                              hipStream_t stream) {
  const float* logits = (const float*)d_in[0];
  const float* pmask  = (const float*)d_in[1];
  const float* gmask  = (const float*)d_in[2];
  const int*   glab   = (const int*)d_in[3];
  const int*   pidx   = (const int*)d_in[4];
  const int*   gidx   = (const int*)d_in[5];
  float* out = (float*)d_out;
  float* ws  = (float*)d_ws;

  const size_t SZ = (size_t)N_*HW_;          // 9,437,184 floats
  float*  ps    = ws;
  float*  bufA  = ws + SZ;
  float*  bufB  = ws + 2*SZ;
  float*  bufC  = ws + 3*SZ;
  float2* lovb  = (float2*)ws;               // reuses [0, 4*SZ) AFTER clDice done
  float*  part  = ws + 4*SZ;                 // [10][64][36]
  float*  red   = part + (size_t)NQTY_*N_*CH_;
  float*  lcls  = red + (size_t)NQTY_*N_;
  float*  lovout= lcls + B_;

  const int gbElem = (int)(SZ/256);                   // 36864
  const int gbLov  = (int)(((size_t)N_*NP2_)/256);    // 65536
  const int gbLovL = (int)(((size_t)N_*NP2_)/LBS_);   // 8192
  const int gbTile = N_ * (H_/32) * (W_/32);          // 9216
  dim3 blk(256);

  // classification + per-pixel base sums
  k_focal<<<B_, 128, 0, stream>>>(logits, glab, pidx, lcls);
  k_sigmoid_fill<<<gbElem, blk, 0, stream>>>(pmask, pidx, ps);
  k_base_stage1<<<N_*CH_, blk, 0, stream>>>(pmask, gmask, pidx, gidx, part);

  // boundary loss
  k_edges<<<gbElem, blk, 0, stream>>>(gmask, gidx, bufA);
  k_bnd_stage1<<<N_*CH_, blk, 0, stream>>>(pmask, gmask, pidx, gidx, bufA, part);

  // soft skeleton of sigmoid(pred): fused erode+open+accumulate per iteration
  k_copy<<<gbElem, blk, 0, stream>>>(ps, bufA);
  k_zero<<<gbElem, blk, 0, stream>>>(bufB);
  {
    float* img = bufA; float* tmp = bufC;
    for (int it=0; it<10; ++it){
      k_skel_iter<<<gbTile, blk, 0, stream>>>(img, tmp, bufB);
      float* s = img; img = tmp; tmp = s;
    }
  }
  k_skel_stage1<<<N_*CH_, blk, 0, stream>>>(bufB, gmask, gidx, (const float*)nullptr, 0,
                                            part + 6*(size_t)N_*CH_);

  // soft skeleton of GT
  k_gather_gm<<<gbElem, blk, 0, stream>>>(gmask, gidx, bufA);
  k_zero<<<gbElem, blk, 0, stream>>>(bufB);
  {
    float* img = bufA; float* tmp = bufC;
    for (int it=0; it<10; ++it){
      k_skel_iter<<<gbTile, blk, 0, stream>>>(img, tmp, bufB);
      float* s = img; img = tmp; tmp = s;
    }
  }
  k_skel_stage1<<<N_*CH_, blk, 0, stream>>>(bufB, gmask, gidx, ps, 1,
                                            part + 8*(size_t)N_*CH_);

  // WMMA reduction of all 10 quantities x 64 instances x 36 chunks
  k_wmma_reduce<<<NQTY_*4, 32, 0, stream>>>(part, red);

  // Lovasz hinge: fill, hybrid global/LDS bitonic sort (descending), scan
  k_lov_fill<<<gbLov, blk, 0, stream>>>(pmask, gmask, pidx, gidx, lovb);
  for (unsigned k = 2; k <= NP2_; k <<= 1){
    for (unsigned j = k >> 1; j >= (unsigned)LBS_; j >>= 1)
      k_bitonic_global<<<gbLov, blk, 0, stream>>>(lovb, k, j);
    k_bitonic_lds<<<gbLovL, blk, 0, stream>>>(lovb, k);
  }
  k_lov_scan<<<N_, blk, 0, stream>>>(lovb, red + 3*N_, lovout);

  // final scalar
  k_combine<<<1, 1, 0, stream>>>(red, lcls, lovout, out);
}